// AttentionGlobalPooling_17600775979467
// MI455X (gfx1250) — compile-verified
//
#include <hip/hip_runtime.h>
#include <hip/hip_bf16.h>

typedef __attribute__((ext_vector_type(2))) float v2f;
typedef __attribute__((ext_vector_type(8))) float v8f;

#define IN_CH   128
#define NGRAPH  1024
#define P2      144   // float2 pitch: 2*144 = 288 ≡ 32 (mod 64) -> half-waves hit disjoint bank halves

// ---------------------------------------------------------------------------
// Kernel 1: scores[i] = relu(x[i,:] @ W1 + b1) . w2 + b2   via f32 WMMA.
// One wave handles 32 nodes (two 16-row A tiles); 8 column tiles of 16;
// 32 K-steps of 4. W1 staged in LDS as (k,k+1)-interleaved float2 so each
// B fragment is one ds_load_b64 into an aligned VGPR pair.
// ---------------------------------------------------------------------------
__global__ __launch_bounds__(256) void k_scores(const float* __restrict__ x,
                                                const float* __restrict__ W1,
                                                const float* __restrict__ b1,
                                                const float* __restrict__ w2,
                                                const float* __restrict__ b2,
                                                float* __restrict__ scores,
                                                int N)
{
    // sWp[p*P2 + n] = { W1[2p][n], W1[2p+1][n] },  p = 0..63, n = 0..127
    __shared__ v2f sWp[64 * P2];     // 72 KB (gfx1250 LDS: up to 320 KB/WG)

    const int tid = threadIdx.x;

    for (int i = tid; i < 64 * IN_CH; i += 256) {
        const int p = i >> 7, n = i & 127;
        v2f pr;
        pr.x = W1[(2 * p) * IN_CH + n];
        pr.y = W1[(2 * p + 1) * IN_CH + n];
        sWp[p * P2 + n] = pr;
    }
    __syncthreads();

    const int wave = tid >> 5;
    const int lane = tid & 31;
    const int hf   = lane >> 4;     // half-wave select
    const int lcol = lane & 15;

    const long long node_base = (long long)blockIdx.x * 256 + wave * 32;

    // A-fragment rows for this lane (clamped: EXEC stays all-ones for WMMA)
    long long r0 = node_base + lcol;
    long long r1 = node_base + 16 + lcol;
    const long long nm1 = (long long)N - 1;
    if (r0 > nm1) r0 = nm1;
    if (r1 > nm1) r1 = nm1;
    const float* xrow0 = x + r0 * IN_CH + hf * 2;  // lanes 0-15: K0,K1 ; 16-31: K2,K3
    const float* xrow1 = x + r1 * IN_CH + hf * 2;

    // 2 x 8 accumulators, seeded with b1 broadcast
    v8f acc0[8], acc1[8];
#pragma unroll
    for (int t = 0; t < 8; ++t) {
        const float bv = b1[t * 16 + lcol];
        acc0[t] = (v8f){bv, bv, bv, bv, bv, bv, bv, bv};
        acc1[t] = (v8f){bv, bv, bv, bv, bv, bv, bv, bv};
    }

    for (int kb = 0; kb < 32; ++kb) {
        const int k0 = kb * 4;
        const v2f a0 = *(const v2f*)(xrow0 + k0);   // global_load_b64
        const v2f a1 = *(const v2f*)(xrow1 + k0);
        const v2f* prow = &sWp[(kb * 2 + hf) * P2]; // pair row kB = k0 + 2*hf
#pragma unroll
        for (int t = 0; t < 8; ++t) {
            const v2f b = prow[t * 16 + lcol];      // one ds_load_b64: {W1[kB][n], W1[kB+1][n]}
            acc0[t] = __builtin_amdgcn_wmma_f32_16x16x4_f32(
                false, a0, false, b, (short)0, acc0[t], false, false);
            acc1[t] = __builtin_amdgcn_wmma_f32_16x16x4_f32(
                false, a1, false, b, (short)0, acc1[t], false, false);
        }
    }

    // relu, weight by w2, fold 8 column tiles into per-row partial scores.
    // C layout: lane holds column (t*16+lcol), rows {0..7} + 8*hf.
    float part0[8] = {0, 0, 0, 0, 0, 0, 0, 0};
    float part1[8] = {0, 0, 0, 0, 0, 0, 0, 0};
#pragma unroll
    for (int t = 0; t < 8; ++t) {
        const float wv = w2[t * 16 + lcol];
#pragma unroll
        for (int e = 0; e < 8; ++e) {
            part0[e] += fmaxf(acc0[t][e], 0.f) * wv;
            part1[e] += fmaxf(acc1[t][e], 0.f) * wv;
        }
    }
#pragma unroll
    for (int m = 1; m < 16; m <<= 1) {
#pragma unroll
        for (int e = 0; e < 8; ++e) {
            part0[e] += __shfl_xor(part0[e], m, 32);
            part1[e] += __shfl_xor(part1[e], m, 32);
        }
    }
    if (lcol == 0) {
        const float bb = b2[0];
#pragma unroll
        for (int e = 0; e < 8; ++e) {
            const long long w0 = node_base + hf * 8 + e;
            const long long w1 = node_base + 16 + hf * 8 + e;
            if (w0 < (long long)N) scores[w0] = part0[e] + bb;
            if (w1 < (long long)N) scores[w1] = part1[e] + bb;
        }
    }
}

// ---------------------------------------------------------------------------
// Kernel 2: segment boundaries via binary search (batch is sorted int64)
// ---------------------------------------------------------------------------
__global__ void k_bounds(const long long* __restrict__ batch, int N,
                         int* __restrict__ seg_start)
{
    const int g = blockIdx.x * blockDim.x + threadIdx.x;
    if (g > NGRAPH) return;
    int lo = 0, hi = N;
    while (lo < hi) {
        const int mid = (lo + hi) >> 1;
        if (batch[mid] < (long long)g) lo = mid + 1; else hi = mid;
    }
    seg_start[g] = lo;
}

// ---------------------------------------------------------------------------
// Kernel 3: per-graph softmax + weighted pooling. One workgroup per graph.
// ---------------------------------------------------------------------------
__global__ __launch_bounds__(256) void k_pool(const float* __restrict__ x,
                                              const float* __restrict__ scores,
                                              const int* __restrict__ seg_start,
                                              float* __restrict__ out, int N)
{
    __shared__ float red[256];
    __shared__ float pool[8][IN_CH];
    __shared__ float sMax, sInv;

    const int g   = blockIdx.x;
    const int s   = seg_start[g];
    const int e   = seg_start[g + 1];
    const int tid = threadIdx.x;

    if (e <= s) {                      // empty graph: segment_sum == 0
        if (tid < IN_CH) out[(long long)g * IN_CH + tid] = 0.f;
        return;
    }

    // --- segment max ---
    float m = -3.402823466e38f;
    for (int i = s + tid; i < e; i += 256) m = fmaxf(m, scores[i]);
    red[tid] = m; __syncthreads();
    for (int st = 128; st > 0; st >>= 1) {
        if (tid < st) red[tid] = fmaxf(red[tid], red[tid + st]);
        __syncthreads();
    }
    if (tid == 0) sMax = red[0];
    __syncthreads();
    const float mx = sMax;

    // --- segment sum of exp ---
    float p = 0.f;
    for (int i = s + tid; i < e; i += 256) p += expf(scores[i] - mx);
    red[tid] = p; __syncthreads();
    for (int st = 128; st > 0; st >>= 1) {
        if (tid < st) red[tid] += red[tid + st];
        __syncthreads();
    }
    if (tid == 0) sInv = 1.f / red[0];
    __syncthreads();
    const float inv = sInv;

    // --- weighted pooling: each wave streams full 512B rows (float4/lane) ---
    const int wave = tid >> 5, lane = tid & 31;
    float4 acc = {0.f, 0.f, 0.f, 0.f};
    for (int i = s + wave; i < e; i += 8) {
        const float a  = expf(scores[i] - mx) * inv;
        const float4 xv = *(const float4*)(x + (long long)i * IN_CH + lane * 4);
        acc.x += xv.x * a; acc.y += xv.y * a;
        acc.z += xv.z * a; acc.w += xv.w * a;
    }
    pool[wave][lane * 4 + 0] = acc.x;
    pool[wave][lane * 4 + 1] = acc.y;
    pool[wave][lane * 4 + 2] = acc.z;
    pool[wave][lane * 4 + 3] = acc.w;
    __syncthreads();
    if (tid < IN_CH) {
        float sum = 0.f;
#pragma unroll
        for (int w = 0; w < 8; ++w) sum += pool[w][tid];
        out[(long long)g * IN_CH + tid] = sum;
    }
}

// ---------------------------------------------------------------------------
extern "C" void kernel_launch(void* const* d_in, const int* in_sizes, int n_in,
                              void* d_out, int out_size, void* d_ws, size_t ws_size,
                              hipStream_t stream)
{
    const float*      x     = (const float*)d_in[0];
    const long long*  batch = (const long long*)d_in[1];
    const float*      W1    = (const float*)d_in[2];
    const float*      b1    = (const float*)d_in[3];
    const float*      w2    = (const float*)d_in[4];
    const float*      b2    = (const float*)d_in[5];
    float*            out   = (float*)d_out;

    const int N = in_sizes[1];                 // node count

    float* scores = (float*)d_ws;
    size_t off = (((size_t)N * sizeof(float)) + 255) & ~(size_t)255;
    int* seg = (int*)((char*)d_ws + off);

    const int nb = (N + 255) / 256;
    k_scores<<<nb, 256, 0, stream>>>(x, W1, b1, w2, b2, scores, N);
    k_bounds<<<(NGRAPH + 1 + 255) / 256, 256, 0, stream>>>(batch, N, seg);
    k_pool<<<NGRAPH, 256, 0, stream>>>(x, scores, seg, out, N);
}